// QuantumAttention_58488864636971
// MI455X (gfx1250) — compile-verified
//
#include <hip/hip_runtime.h>

typedef __attribute__((ext_vector_type(2))) float v2f;
typedef __attribute__((ext_vector_type(8))) float v8f;

// ---------------------------------------------------------------------------
// Setup: build U1 = sigma1 * (R0(x)R1(x)R2(x)R3), U2 = sigma2 * (R4..R7)
// as 16x16 complex matrices (re/im planes) into d_ws.
// ws layout: [0..256) U1re, [256..512) U1im, [512..768) U2re, [768..1024) U2im
// Rot(phi,theta,omega) = [[e^{-i(p+o)/2} c, -e^{+i(p-o)/2} s],
//                         [e^{-i(p-o)/2} s,  e^{+i(p+o)/2} c]]
// ---------------------------------------------------------------------------
__global__ void qa_setup(const float* __restrict__ w, float* __restrict__ U) {
  int t = blockIdx.x * blockDim.x + threadIdx.x;
  if (t >= 256) return;
  int i = t >> 4;    // row index (bits s0..s3, s0 = MSB)
  int ip = t & 15;   // column index
  float re[2] = {1.f, 1.f}, im[2] = {0.f, 0.f};
  for (int reg = 0; reg < 2; ++reg) {
    for (int m = 0; m < 4; ++m) {
      int wire = reg * 4 + m;
      float phi = w[wire * 3 + 0];
      float th  = w[wire * 3 + 1];
      float om  = w[wire * 3 + 2];
      float c = cosf(0.5f * th), sn = sinf(0.5f * th);
      int s  = (i  >> (3 - m)) & 1;   // output bit for this wire
      int sp = (ip >> (3 - m)) & 1;   // input bit
      float ang, mag;
      if (s == 0 && sp == 0)      { ang = -0.5f * (phi + om); mag =  c;  }
      else if (s == 0 && sp == 1) { ang =  0.5f * (phi - om); mag = -sn; }
      else if (s == 1 && sp == 0) { ang = -0.5f * (phi - om); mag =  sn; }
      else                        { ang =  0.5f * (phi + om); mag =  c;  }
      float rre = mag * cosf(ang), rim = mag * sinf(ang);
      float nre = re[reg] * rre - im[reg] * rim;
      float nim = re[reg] * rim + im[reg] * rre;
      re[reg] = nre; im[reg] = nim;
    }
  }
  // intra-register CZ signs: (-1)^{s0 s1 + s1 s2 + s2 s3}, row-dependent only
  int b3 = (i >> 3) & 1, b2 = (i >> 2) & 1, b1 = (i >> 1) & 1, b0 = i & 1;
  float sg = (((b3 & b2) ^ (b2 & b1) ^ (b1 & b0)) != 0) ? -1.f : 1.f;
  U[t]        = re[0] * sg;
  U[256 + t]  = im[0] * sg;
  U[512 + t]  = re[1] * sg;
  U[768 + t]  = im[1] * sg;
}

// ---------------------------------------------------------------------------
// Main fused kernel: 1 wave handles a tile of 16 batch elements.
//  - WMMA f32 16x16x4 (K=16 via 4 chained ops) computes G1re/G1im/G2re/G2im
//  - stage D accumulators to LDS
//  - per element: build psi[i,j] = g1[i]*g2[j]*crossSign, 4 symmetrization
//    passes (I+SWAP)/2 via shfl_xor, reduce |psi|^2, write out.
// psi layout: idx8 = i*16 + j (i,j in [0,16)); lane = idx8>>3, slot = idx8&7.
// ---------------------------------------------------------------------------
__global__ __launch_bounds__(128) void qa_main(const float* __restrict__ x,
                                               const float* __restrict__ U,
                                               float* __restrict__ out, int B) {
  __shared__ float G[4][4][256];  // [wave][{g1re,g1im,g2re,g2im}][m*16+n]
  const int lane = threadIdx.x & 31;
  const int wv   = threadIdx.x >> 5;
  const int tile = blockIdx.x * 4 + wv;
  const int b0   = tile * 16;
  const bool active = (b0 < B);

  const int h    = lane >> 4;   // half-wave selector (K pairing)
  const int mrow = lane & 15;   // A: batch row within tile; B: column N

  if (active) {
    const float* a1p  = x + (size_t)(b0 + mrow) * 32;       // f1 row
    const float* a2p  = a1p + 16;                           // f2 row
    const float* r1re = U + mrow * 16;
    const float* r1im = U + 256 + mrow * 16;
    const float* r2re = U + 512 + mrow * 16;
    const float* r2im = U + 768 + mrow * 16;

    v8f c1re = {}, c1im = {}, c2re = {}, c2im = {};
#pragma unroll
    for (int kk = 0; kk < 4; ++kk) {
      const int k0 = 4 * kk + 2 * h;
      v2f a1, a2, b1re, b1im, b2re, b2im;
      a1[0] = a1p[k0];  a1[1] = a1p[k0 + 1];
      a2[0] = a2p[k0];  a2[1] = a2p[k0 + 1];
      b1re[0] = r1re[k0]; b1re[1] = r1re[k0 + 1];
      b1im[0] = r1im[k0]; b1im[1] = r1im[k0 + 1];
      b2re[0] = r2re[k0]; b2re[1] = r2re[k0 + 1];
      b2im[0] = r2im[k0]; b2im[1] = r2im[k0 + 1];
      c1re = __builtin_amdgcn_wmma_f32_16x16x4_f32(false, a1, false, b1re,
                                                   (short)0, c1re, false, false);
      c1im = __builtin_amdgcn_wmma_f32_16x16x4_f32(false, a1, false, b1im,
                                                   (short)0, c1im, false, false);
      c2re = __builtin_amdgcn_wmma_f32_16x16x4_f32(false, a2, false, b2re,
                                                   (short)0, c2re, false, false);
      c2im = __builtin_amdgcn_wmma_f32_16x16x4_f32(false, a2, false, b2im,
                                                   (short)0, c2im, false, false);
    }
    // stage accumulators: D[M = r + 8h][N = mrow]
#pragma unroll
    for (int r = 0; r < 8; ++r) {
      const int idx = (r + 8 * h) * 16 + mrow;
      G[wv][0][idx] = c1re[r];
      G[wv][1][idx] = c1im[r];
      G[wv][2][idx] = c2re[r];
      G[wv][3][idx] = c2im[r];
    }
  }
  __syncthreads();

  if (active) {
    const int iidx  = lane >> 1;          // i (4 bits), constant per lane
    const int jb    = (lane & 1) << 3;    // high bit of j
    const int i_lsb = iidx & 1;
    const int i_msb = (iidx >> 3) & 1;

    for (int m = 0; m < 16; ++m) {
      const float g1re = G[wv][0][m * 16 + iidx];
      const float g1im = G[wv][1][m * 16 + iidx];
      float pre[8], pim[8];
#pragma unroll
      for (int r = 0; r < 8; ++r) {
        const int j = jb + r;
        const float g2re = G[wv][2][m * 16 + j];
        const float g2im = G[wv][3][m * 16 + j];
        float vre = g1re * g2re - g1im * g2im;
        float vim = g1re * g2im + g1im * g2re;
        // cross CZ(3,4) & CZ(7,0): (-1)^{lsb(i)*msb(j) + msb(i)*lsb(j)}
        const int sgn = (i_lsb & (j >> 3)) ^ (i_msb & (j & 1));
        pre[r] = sgn ? -vre : vre;
        pim[r] = sgn ? -vim : vim;
      }
      // pass 3: swap i3<->j3  (lane bits 4 and 0)
      {
        const int cond = ((lane >> 4) ^ lane) & 1;
#pragma unroll
        for (int r = 0; r < 8; ++r) {
          const float qre = __shfl_xor(pre[r], 17, 32);
          const float qim = __shfl_xor(pim[r], 17, 32);
          if (cond) { pre[r] = 0.5f * (pre[r] + qre); pim[r] = 0.5f * (pim[r] + qim); }
        }
      }
      // pass 2: swap i2<->j2  (lane bit 3 <-> slot bit 2)
      {
        float tre[8], tim[8];
#pragma unroll
        for (int r = 0; r < 8; ++r) {
          const float qre = __shfl_xor(pre[r ^ 4], 8, 32);
          const float qim = __shfl_xor(pim[r ^ 4], 8, 32);
          const int cond = ((lane >> 3) ^ (r >> 2)) & 1;
          tre[r] = cond ? 0.5f * (pre[r] + qre) : pre[r];
          tim[r] = cond ? 0.5f * (pim[r] + qim) : pim[r];
        }
#pragma unroll
        for (int r = 0; r < 8; ++r) { pre[r] = tre[r]; pim[r] = tim[r]; }
      }
      // pass 1: swap i1<->j1  (lane bit 2 <-> slot bit 1)
      {
        float tre[8], tim[8];
#pragma unroll
        for (int r = 0; r < 8; ++r) {
          const float qre = __shfl_xor(pre[r ^ 2], 4, 32);
          const float qim = __shfl_xor(pim[r ^ 2], 4, 32);
          const int cond = ((lane >> 2) ^ (r >> 1)) & 1;
          tre[r] = cond ? 0.5f * (pre[r] + qre) : pre[r];
          tim[r] = cond ? 0.5f * (pim[r] + qim) : pim[r];
        }
#pragma unroll
        for (int r = 0; r < 8; ++r) { pre[r] = tre[r]; pim[r] = tim[r]; }
      }
      // pass 0: swap i0<->j0  (lane bit 1 <-> slot bit 0)
      {
        float tre[8], tim[8];
#pragma unroll
        for (int r = 0; r < 8; ++r) {
          const float qre = __shfl_xor(pre[r ^ 1], 2, 32);
          const float qim = __shfl_xor(pim[r ^ 1], 2, 32);
          const int cond = ((lane >> 1) ^ r) & 1;
          tre[r] = cond ? 0.5f * (pre[r] + qre) : pre[r];
          tim[r] = cond ? 0.5f * (pim[r] + qim) : pim[r];
        }
#pragma unroll
        for (int r = 0; r < 8; ++r) { pre[r] = tre[r]; pim[r] = tim[r]; }
      }
      // ||P psi||^2
      float sm = 0.f;
#pragma unroll
      for (int r = 0; r < 8; ++r) sm += pre[r] * pre[r] + pim[r] * pim[r];
      sm += __shfl_xor(sm, 16, 32);
      sm += __shfl_xor(sm, 8, 32);
      sm += __shfl_xor(sm, 4, 32);
      sm += __shfl_xor(sm, 2, 32);
      sm += __shfl_xor(sm, 1, 32);
      if (lane == m) out[b0 + m] = sm;
    }
  }
}

extern "C" void kernel_launch(void* const* d_in, const int* in_sizes, int n_in,
                              void* d_out, int out_size, void* d_ws, size_t ws_size,
                              hipStream_t stream) {
  (void)n_in; (void)out_size; (void)ws_size;
  const float* x = (const float*)d_in[0];      // (B, 32) normalized features
  const float* w = (const float*)d_in[1];      // (1, 8, 3) weights
  float* out = (float*)d_out;                  // (B,)
  float* U = (float*)d_ws;                     // 1024 floats of scratch
  const int B = in_sizes[0] / 32;

  qa_setup<<<1, 256, 0, stream>>>(w, U);
  const int tiles  = (B + 15) / 16;
  const int blocks = (tiles + 3) / 4;
  qa_main<<<blocks, 128, 0, stream>>>(x, U, out, B);
}